// KernelDynamics_28080496181743
// MI455X (gfx1250) — compile-verified
//
#include <hip/hip_runtime.h>

#define N_PART   512
#define N_DIMS   3
#define NK       32
#define NT       10
#define NBATCH   8
#define EPSF     1e-6f
#define ITILES   (N_PART / 16)   // 32

typedef __attribute__((ext_vector_type(2))) float v2f;
typedef __attribute__((ext_vector_type(8))) float v8f;

// ---------------------------------------------------------------------------
// Prep: collapse the T dimension.  trbfs(t) -> w_eff[32], c0.  ws[0..31]=w_eff,
// ws[32]=c0.  Single block, serial where it matters -> deterministic.
// ---------------------------------------------------------------------------
__global__ void prep_kernel(const float* __restrict__ t,
                            const float* __restrict__ mus_t,
                            const float* __restrict__ nlg_t,
                            const float* __restrict__ weights,
                            const float* __restrict__ bias,
                            const float* __restrict__ imp,
                            float* __restrict__ ws) {
  __shared__ float s_trbf[NT];
  __shared__ float s_weff[NK];
  const int tid = threadIdx.x;
  if (tid == 0) {
    float tv = t[0];
    float e[NT];
    float s = 0.f;
    for (int k = 0; k < NT; ++k) {
      float ig = __expf(2.f * nlg_t[k]);      // inv_gamma2 = exp(nlg)^2
      float df = tv - mus_t[k];
      e[k] = __expf(-df * df * ig);
      s += e[k];
    }
    float inv = 1.f / (EPSF + s);
    for (int k = 0; k < NT; ++k) s_trbf[k] = e[k] * inv;
  }
  __syncthreads();
  if (tid < NK) {
    float acc = 0.f;
    for (int k = 0; k < NT; ++k) acc += weights[tid * NT + k] * s_trbf[k];
    s_weff[tid] = acc;
    ws[tid] = acc;
  }
  __syncthreads();
  if (tid == 0) {
    float c0 = 0.f;
    for (int k = 0; k < NT; ++k) c0 += bias[k] * s_trbf[k];
    for (int k = 0; k < NK; ++k) c0 += imp[k] * imp[k] * s_weff[k];
    ws[NK] = c0;
  }
}

// ---------------------------------------------------------------------------
// Main pair kernel.  One wave (32 threads) per block.
//   block = ((b*32 + itile)*JC + jc)
//   lane L handles pair p = L&15 (particle i = itile*16+p), k%4 in {0,1} for
//   L<16 and {2,3} for L>=16.
// WMMA f32 16x16x4, transposed use:
//   A[m][k] = (m even) ? w_eff[k] : 1.0   (constant, preloaded)
//   B[k][n] = rbf(pair n, k)  (C1)  /  drbf(pair n, k)  (C2)
//   => C VGPR0 = sw(pair)   at lane = pair (both half-waves)
//      C VGPR1 = srbf(pair)
// ---------------------------------------------------------------------------
__global__ __launch_bounds__(32) void pair_kernel(const float* __restrict__ x,
                                                  const float* __restrict__ mus,
                                                  const float* __restrict__ nlg,
                                                  const float* __restrict__ ws,
                                                  float* __restrict__ part,
                                                  int JC) {
  const int lane  = threadIdx.x;
  const int p     = lane & 15;
  const int blk   = blockIdx.x;
  const int jc    = blk % JC;
  const int bt    = blk / JC;           // b*32 + itile
  const int itile = bt & (ITILES - 1);
  const int b     = bt / ITILES;
  const int i     = itile * 16 + p;

  const float* xb = x + (size_t)b * (N_PART * N_DIMS);
  const float xi0 = xb[i * 3 + 0];
  const float xi1 = xb[i * 3 + 1];
  const float xi2 = xb[i * 3 + 2];

  const float c0 = ws[NK];
  const int   kh = (lane < 16) ? 0 : 2; // which k%4 half this lane owns
  const float LOG2E = 1.4426950408889634f;
  const bool  wrow = ((p & 1) == 0);    // even A-rows carry w_eff, odd carry 1

  float mu0[8], mu1[8], gl0[8], gl1[8], gm0[8], gm1[8];
  v2f a[8];
#pragma unroll
  for (int s = 0; s < 8; ++s) {
    const int k0 = 4 * s + kh, k1 = k0 + 1;
    mu0[s] = mus[k0];
    mu1[s] = mus[k1];
    const float ig0 = __expf(2.f * nlg[k0]);
    const float ig1 = __expf(2.f * nlg[k1]);
    gl0[s] = ig0 * LOG2E;               // for exp2
    gl1[s] = ig1 * LOG2E;
    gm0[s] = -2.f * ig0;                // drbf factor
    gm1[s] = -2.f * ig1;
    a[s].x = wrow ? ws[k0] : 1.0f;
    a[s].y = wrow ? ws[k1] : 1.0f;
  }

  float fx = 0.f, fy = 0.f, fz = 0.f, dv = 0.f;
  const int jlen = N_PART / JC;
  const int j0   = jc * jlen;

  for (int j = j0; j < j0 + jlen; ++j) {
    const float xj0 = xb[j * 3 + 0];
    const float xj1 = xb[j * 3 + 1];
    const float xj2 = xb[j * 3 + 2];
    const float rx = xi0 - xj0, ry = xi1 - xj1, rz = xi2 - xj2;
    const float d = sqrtf(rx * rx + ry * ry + rz * rz + EPSF);

    v8f C1 = {0.f, 0.f, 0.f, 0.f, 0.f, 0.f, 0.f, 0.f};
    v8f C2 = {0.f, 0.f, 0.f, 0.f, 0.f, 0.f, 0.f, 0.f};
#pragma unroll
    for (int s = 0; s < 8; ++s) {
      const float d0 = d - mu0[s];
      const float d1 = d - mu1[s];
      const float e0 = __builtin_amdgcn_exp2f(-(d0 * d0) * gl0[s]);
      const float e1 = __builtin_amdgcn_exp2f(-(d1 * d1) * gl1[s]);
      v2f brbf  = {e0, e1};
      v2f bdrbf = {gm0[s] * d0 * e0, gm1[s] * d1 * e1};
      C1 = __builtin_amdgcn_wmma_f32_16x16x4_f32(false, a[s], false, brbf,
                                                 (short)0, C1, false, false);
      C2 = __builtin_amdgcn_wmma_f32_16x16x4_f32(false, a[s], false, bdrbf,
                                                 (short)0, C2, false, false);
    }
    const float sw    = C1[0];
    const float srbf  = C1[1];
    const float sdw   = C2[0];
    const float sdrbf = C2[1];
    const float inv = 1.0f / (EPSF + srbf);
    const float fm  = sw * inv + c0;
    const float dfm = (sdw - sw * sdrbf * inv) * inv;
    if (j != i) {                       // mask: remove diagonal
      fx += rx * fm;
      fy += ry * fm;
      fz += rz * fm;
      dv += d * dfm + 3.0f * fm;
    }
  }

  // Each pair is replicated in both half-waves: reduce all 32 lanes, halve.
  for (int off = 16; off >= 1; off >>= 1) dv += __shfl_xor(dv, off, 32);

  float* bw = part + (size_t)blk * 49;
  if (lane < 16) {
    bw[p * 3 + 0] = fx;
    bw[p * 3 + 1] = fy;
    bw[p * 3 + 2] = fz;
  }
  if (lane == 0) bw[48] = 0.5f * dv;
}

// ---------------------------------------------------------------------------
// Deterministic fixed-order reduction of per-block partials into d_out.
// out[0 .. 12287] = forces, out[12288 .. 12295] = -divergence.
// ---------------------------------------------------------------------------
__global__ void reduce_kernel(const float* __restrict__ part,
                              float* __restrict__ out, int JC) {
  const int e = blockIdx.x * blockDim.x + threadIdx.x;
  const int total = NBATCH * N_PART * N_DIMS;
  if (e < total) {
    const int b   = e / (N_PART * N_DIMS);
    const int rem = e % (N_PART * N_DIMS);
    const int i   = rem / 3;
    const int c   = rem % 3;
    const int itile = i >> 4;
    const int p     = i & 15;
    float acc = 0.f;
    for (int jc = 0; jc < JC; ++jc) {
      const int blk = (b * ITILES + itile) * JC + jc;
      acc += part[(size_t)blk * 49 + p * 3 + c];
    }
    out[e] = acc;
  } else if (e < total + NBATCH) {
    const int b = e - total;
    float acc = 0.f;
    for (int itile = 0; itile < ITILES; ++itile)
      for (int jc = 0; jc < JC; ++jc)
        acc += part[(size_t)((b * ITILES + itile) * JC + jc) * 49 + 48];
    out[total + b] = -acc;              // reference returns -divergence
  }
}

// ---------------------------------------------------------------------------
extern "C" void kernel_launch(void* const* d_in, const int* in_sizes, int n_in,
                              void* d_out, int out_size, void* d_ws,
                              size_t ws_size, hipStream_t stream) {
  (void)in_sizes; (void)n_in; (void)out_size;
  const float* t       = (const float*)d_in[0];
  const float* x       = (const float*)d_in[1];
  const float* mus     = (const float*)d_in[2];
  const float* nlg     = (const float*)d_in[3];
  const float* mus_t   = (const float*)d_in[4];
  const float* nlg_t   = (const float*)d_in[5];
  const float* weights = (const float*)d_in[6];
  const float* bias    = (const float*)d_in[7];
  const float* imp     = (const float*)d_in[8];
  float* out = (float*)d_out;
  float* ws  = (float*)d_ws;

  // j-chunking for parallelism; scratch: 64 header floats + 49 per block.
  int JC = 4;
  size_t need = (size_t)(64 + NBATCH * ITILES * JC * 49) * sizeof(float);
  if (ws_size < need) JC = 1;           // conservative fallback (~50 KB)

  prep_kernel<<<1, 64, 0, stream>>>(t, mus_t, nlg_t, weights, bias, imp, ws);

  const int nblk = NBATCH * ITILES * JC;
  pair_kernel<<<nblk, 32, 0, stream>>>(x, mus, nlg, ws, ws + 64, JC);

  const int tot = NBATCH * N_PART * N_DIMS + NBATCH;
  reduce_kernel<<<(tot + 255) / 256, 256, 0, stream>>>(ws + 64, out, JC);
}